// Model_5540507812360
// MI455X (gfx1250) — compile-verified
//
#include <hip/hip_runtime.h>
#include <math.h>

typedef float v2f __attribute__((ext_vector_type(2)));
typedef float v8f __attribute__((ext_vector_type(8)));

#define BB    16384
#define DD    65
#define NAA   64
#define NTOT  129   // D + NA

// workspace layout (float offsets)
#define OFF_TWP   0         // 68*64 padded thresh_w^T  [k][n]
#define OFF_WC    4352      // 128*64 combined [C_xt ; S] [k][n]
#define OFF_CXY   12544     // 64
#define OFF_CYT   12608     // 64
#define OFF_LT    12672     // 64
#define OFF_LY    12736     // 1
#define OFF_PINT  12800     // 256 per-block partials
#define OFF_PNEUT 13056     // 256
#define OFF_TD    16384     // B*64
#define OFF_TS    1064960   // B*64  (16384 + 1048576)

__device__ __forceinline__ int pair_idx(int i, int j) {
  // i < j, N = 129; index of pair (i,j) inside main_w (after the N linear terms)
  return NTOT + (i * (2 * NTOT - 1 - i)) / 2 + (j - i - 1);
}

__device__ __forceinline__ v8f wmma_f32(v2f a, v2f b, v8f c) {
  return __builtin_amdgcn_wmma_f32_16x16x4_f32(false, a, false, b, (short)0, c,
                                               false, false);
}

// ---------------------------------------------------------------- prep
__global__ void prep_kernel(const float* __restrict__ thresh_w,
                            const float* __restrict__ main_w,
                            float* __restrict__ ws) {
  float* TWp = ws + OFF_TWP;
  float* Wc  = ws + OFF_WC;
  // padded thresh_w^T : TWp[k][n], k in [0,68), rows 65..67 zero
  for (int idx = threadIdx.x; idx < 68 * 64; idx += blockDim.x) {
    int k = idx >> 6, n = idx & 63;
    TWp[idx] = (k < DD) ? thresh_w[n * DD + k] : 0.0f;
  }
  // combined A-side weights: rows 0..63 = C_xt (x_i vs t_j pairs),
  // rows 64..127 = S = 0.5*(U + U^T) from t-t pairs
  for (int idx = threadIdx.x; idx < 128 * 64; idx += blockDim.x) {
    int i = idx >> 6, j = idx & 63;
    float v;
    if (i < 64) {
      v = main_w[pair_idx(i, 65 + j)];
    } else {
      int ii = i - 64;
      if (ii == j) v = 0.0f;
      else {
        int lo = ii < j ? ii : j;
        int hi = ii < j ? j : ii;
        v = 0.5f * main_w[pair_idx(65 + lo, 65 + hi)];
      }
    }
    Wc[idx] = v;
  }
  if (threadIdx.x < 64) {
    int j = threadIdx.x;
    ws[OFF_CXY + j] = main_w[pair_idx(j, 64)];       // x_j * y pair coeff
    ws[OFF_CYT + j] = main_w[pair_idx(64, 65 + j)];  // y * t_j pair coeff
    ws[OFF_LT  + j] = main_w[65 + j];                // linear t_j coeff
  }
  if (threadIdx.x == 0) ws[OFF_LY] = main_w[64];     // linear y coeff
}

// ------------------------------------------------- GEMM1 + tanh + penalties
// block = 128 threads (4 waves); each wave: 16 rows x 64 cols; block: 64 rows
__global__ void gemm1_kernel(const float* __restrict__ inps,
                             const float* __restrict__ thresh_w,
                             const float* __restrict__ thresh_b,
                             float* __restrict__ ws) {
  const float* TWp = ws + OFF_TWP;
  float* TD = ws + OFF_TD;
  float* TS = ws + OFF_TS;
  __shared__ float sy[64];
  __shared__ float sred[128];

  const int tid   = threadIdx.x;
  const int rows0 = blockIdx.x * 64;
  if (tid < 64) sy[tid] = inps[(rows0 + tid) * DD + 64];  // y per row
  __syncthreads();

  const int wid  = tid >> 5;
  const int lane = tid & 31;
  const int half = lane >> 4;   // A: K pair select / C: M bias
  const int ln   = lane & 15;   // A: M row / B,C: N column
  const int row0 = rows0 + wid * 16;

  v8f acc[4] = {};
  const float* arow = inps + (row0 + ln) * DD;
  const int kab = 2 * half;

  for (int k0 = 0; k0 < 64; k0 += 4) {
    const int ka = k0 + kab;
    v2f a; a[0] = arow[ka]; a[1] = arow[ka + 1];
#pragma unroll
    for (int t = 0; t < 4; ++t) {
      v2f b;
      b[0] = TWp[ka * 64 + t * 16 + ln];
      b[1] = TWp[(ka + 1) * 64 + t * 16 + ln];
      acc[t] = wmma_f32(a, b, acc[t]);
    }
  }
  {  // tail K-step: k = 64..67, only k==64 (== y column) is real
    const int ka = 64 + kab;
    v2f a;
    a[0] = (ka < DD) ? arow[ka] : 0.0f;
    a[1] = 0.0f;
#pragma unroll
    for (int t = 0; t < 4; ++t) {
      v2f b;
      b[0] = TWp[ka * 64 + t * 16 + ln];
      b[1] = TWp[(ka + 1) * 64 + t * 16 + ln];
      acc[t] = wmma_f32(a, b, acc[t]);
    }
  }

  // epilogue: w1 -> w2 -> tanh -> td/ts stores + penalty accumulation
  float pint = 0.0f, pneut = 0.0f;
#pragma unroll
  for (int t = 0; t < 4; ++t) {
    const int n   = t * 16 + ln;
    const float bn  = thresh_b[n];
    const float twy = thresh_w[n * DD + 64];
#pragma unroll
    for (int v = 0; v < 8; ++v) {
      const int m = v + 8 * half;
      const int r = row0 + m;
      const float yv = sy[wid * 16 + m];
      const float w1 = acc[t][v] + bn;
      const float w2 = w1 - 2.0f * yv * twy;
      const float t1 = tanhf(w1);
      const float t2 = tanhf(w2);
      TD[r * 64 + n] = t1 - t2;
      TS[r * 64 + n] = t1 + t2;
      const float a1 = 1.0f - t1 * t1;
      const float a2 = 1.0f - t2 * t2;
      pint += (1.0f / 300.0f) * (a1 * a1 + a2 * a2);
      const float nn = t2 * w2 - t1 * w1;
      pneut += nn * nn;
    }
  }

  // deterministic block reductions
  sred[tid] = pint; __syncthreads();
  for (int s = 64; s > 0; s >>= 1) {
    if (tid < s) sred[tid] += sred[tid + s];
    __syncthreads();
  }
  if (tid == 0) ws[OFF_PINT + blockIdx.x] = sred[0];
  __syncthreads();
  sred[tid] = pneut; __syncthreads();
  for (int s = 64; s > 0; s >>= 1) {
    if (tid < s) sred[tid] += sred[tid + s];
    __syncthreads();
  }
  if (tid == 0) ws[OFF_PNEUT + blockIdx.x] = sred[0];
}

// ------------------------------------------------- GEMM2 + diff finalize
__global__ void gemm2_kernel(const float* __restrict__ inps,
                             const float* __restrict__ ws,
                             float* __restrict__ out) {
  const float* Wc  = ws + OFF_WC;
  const float* Cxy = ws + OFF_CXY;
  const float* Cyt = ws + OFF_CYT;
  const float* Lt  = ws + OFF_LT;
  const float* TD  = ws + OFF_TD;
  const float* TS  = ws + OFF_TS;
  __shared__ float sy[64];
  __shared__ float sxd[64];

  const int tid   = threadIdx.x;
  const int rows0 = blockIdx.x * 64;
  const int wid   = tid >> 5;
  const int lane  = tid & 31;
  const int half  = lane >> 4;
  const int ln    = lane & 15;
  const int row0  = rows0 + wid * 16;

  if (tid < 64) sy[tid] = inps[(rows0 + tid) * DD + 64];

  // per-row xdot = x . C_xy  (each half does 32 MACs, combine across halves)
  {
    const float* xr = inps + (row0 + ln) * DD + 32 * half;
    const float* cw = Cxy + 32 * half;
    float p = 0.0f;
#pragma unroll 8
    for (int i = 0; i < 32; ++i) p += xr[i] * cw[i];
    p += __shfl_xor(p, 16, 32);
    if (half == 0) sxd[wid * 16 + ln] = p;
  }
  __syncthreads();

  v8f acc[4] = {};
  const float* arow_x = inps + (row0 + ln) * DD;
  const float* arow_t = TS + (row0 + ln) * 64;
  const int kab = 2 * half;

  for (int k0 = 0; k0 < 128; k0 += 4) {
    const int ka = k0 + kab;
    v2f a;
    if (k0 < 64) { a[0] = arow_x[ka];      a[1] = arow_x[ka + 1]; }
    else         { a[0] = arow_t[ka - 64]; a[1] = arow_t[ka - 63]; }
#pragma unroll
    for (int t = 0; t < 4; ++t) {
      v2f b;
      b[0] = Wc[ka * 64 + t * 16 + ln];
      b[1] = Wc[(ka + 1) * 64 + t * 16 + ln];
      acc[t] = wmma_f32(a, b, acc[t]);
    }
  }

  // per-element: (Lt[n] + g[m][n]) * td + y*Cyt[n]*ts ; row-sum over n
  float rsum[8] = {0, 0, 0, 0, 0, 0, 0, 0};
#pragma unroll
  for (int t = 0; t < 4; ++t) {
    const int n = t * 16 + ln;
    const float lt  = Lt[n];
    const float cyt = Cyt[n];
#pragma unroll
    for (int v = 0; v < 8; ++v) {
      const int m = v + 8 * half;
      const int r = row0 + m;
      const float td = TD[r * 64 + n];
      const float ts = TS[r * 64 + n];
      rsum[v] += (lt + acc[t][v]) * td + sy[wid * 16 + m] * cyt * ts;
    }
  }
#pragma unroll
  for (int v = 0; v < 8; ++v) {
    rsum[v] += __shfl_xor(rsum[v], 1, 32);
    rsum[v] += __shfl_xor(rsum[v], 2, 32);
    rsum[v] += __shfl_xor(rsum[v], 4, 32);
    rsum[v] += __shfl_xor(rsum[v], 8, 32);
  }
  if (ln == 0) {
    const float Ly = ws[OFF_LY];
#pragma unroll
    for (int v = 0; v < 8; ++v) {
      const int m = v + 8 * half;
      const int r = row0 + m;
      const float yv = sy[wid * 16 + m];
      out[r] = 1.0f + 2.0f * yv * (Ly + sxd[wid * 16 + m]) + rsum[v];
    }
  }
}

// ------------------------------------------------- final deterministic sums
__global__ void finalize_kernel(const float* __restrict__ ws,
                                float* __restrict__ out) {
  if (threadIdx.x == 0) {
    float s1 = 0.0f, s2 = 0.0f;
    for (int i = 0; i < 256; ++i) {
      s1 += ws[OFF_PINT + i];
      s2 += ws[OFF_PNEUT + i];
    }
    out[BB]     = s1;  // int_penalty
    out[BB + 1] = s2;  // neut_penalty
  }
}

extern "C" void kernel_launch(void* const* d_in, const int* in_sizes, int n_in,
                              void* d_out, int out_size, void* d_ws,
                              size_t ws_size, hipStream_t stream) {
  (void)in_sizes; (void)n_in; (void)out_size; (void)ws_size;
  const float* inps     = (const float*)d_in[0];
  const float* thresh_w = (const float*)d_in[1];
  const float* thresh_b = (const float*)d_in[2];
  const float* main_w   = (const float*)d_in[3];
  float* out = (float*)d_out;
  float* ws  = (float*)d_ws;

  prep_kernel<<<1, 256, 0, stream>>>(thresh_w, main_w, ws);
  gemm1_kernel<<<BB / 64, 128, 0, stream>>>(inps, thresh_w, thresh_b, ws);
  gemm2_kernel<<<BB / 64, 128, 0, stream>>>(inps, ws, out);
  finalize_kernel<<<1, 64, 0, stream>>>(ws, out);
}